// Quantizer_56307021250938
// MI455X (gfx1250) — compile-verified
//
#include <hip/hip_runtime.h>
#include <hip/hip_bf16.h>

typedef __attribute__((ext_vector_type(2))) float v2f;
typedef __attribute__((ext_vector_type(8))) float v8f;
typedef __attribute__((ext_vector_type(4))) unsigned int v4u;
typedef __attribute__((ext_vector_type(4))) int v4i;
typedef __attribute__((ext_vector_type(8))) int v8i;

#define NUM_EMB   512
#define EMB_DIM   64
#define HW        256           // 16*16 pixels per image
#define IMG_ELEMS (EMB_DIM*HW)  // 16384 floats per image
#define ZQ_ELEMS  (1024*IMG_ELEMS)

#define TILE_W    128           // pixels per TDM phase
#define ROW_PITCH 136           // LDS row pitch in floats (128 + 8 TDM pad dwords)
                                // 2*136 % 64 == 16 -> half-waves hit disjoint banks

// ---------------------------------------------------------------------------
// Kernel 0: zero the bincount accumulator and precompute ||e_k||^2
// ---------------------------------------------------------------------------
__global__ void vq_prep_kernel(const float* __restrict__ e,
                               float* __restrict__ e2,
                               unsigned int* __restrict__ n_i) {
    int k = blockIdx.x * blockDim.x + threadIdx.x;
    if (k < NUM_EMB) {
        n_i[k] = 0u;
        const float4* er = (const float4*)(e + (size_t)k * EMB_DIM);
        float s = 0.0f;
#pragma unroll
        for (int i = 0; i < EMB_DIM / 4; ++i) {
            float4 v = er[i];
            s += v.x * v.x + v.y * v.y + v.z * v.z + v.w * v.w;
        }
        e2[k] = s;
    }
}

// ---------------------------------------------------------------------------
// Kernel 1: per-image nearest-code search (WMMA f32 + TDM), gather, histogram.
// One block per image b, 256 threads = 8 waves.
// Two phases; each phase TDM-loads z_e[b, :, p*128 : p*128+128] (64 rows x
// 128 px, row-contiguous) into LDS with 8-dword row padding, then each wave
// computes one 16-row tile against all 512 codes:
//   dist'[m,k] = ||e_k||^2 - 2 z_m.e_k  via 16x v_wmma_f32_16x16x4_f32.
// ---------------------------------------------------------------------------
__global__ void __launch_bounds__(256)
vq_main_kernel(const float* __restrict__ z_e,
               const float* __restrict__ e,
               const float* __restrict__ e2,
               float* __restrict__ z_q,
               unsigned int* __restrict__ n_i) {
    __shared__ float          z_lds[EMB_DIM * ROW_PITCH];  // 34,816 B
    __shared__ unsigned int   hist[NUM_EMB];               //  2,048 B
    __shared__ unsigned short idx_s[HW];                   //    512 B

    const int b   = blockIdx.x;
    const int tid = threadIdx.x;

    hist[tid]       = 0u;
    hist[tid + 256] = 0u;
    __syncthreads();

    const int wave = tid >> 5;
    const int lane = tid & 31;
    const int l16  = lane & 15;   // column N within a 16-wide tile (C layout)
    const int lh   = lane >> 4;   // half-wave select (K interleave for A/B)

    const float*   zb       = z_e + (size_t)b * IMG_ELEMS;
    const unsigned lds_base = (unsigned)(size_t)(void*)z_lds;

    for (int p = 0; p < 2; ++p) {
        // ---- TDM: async DMA half-image into LDS (wave 0 issues) ----------
        if (wave == 0) {
            unsigned long long ga = (unsigned long long)(const void*)(zb + p * TILE_W);
            v4u g0;
            g0[0] = 1u;                                   // count=1, user D#
            g0[1] = lds_base;                             // lds_addr (bytes)
            g0[2] = (unsigned)(ga & 0xffffffffu);         // global_addr[31:0]
            g0[3] = (unsigned)(ga >> 32) | 0x80000000u;   // addr[56:32] | type=2

            v8i g1;
            // data_size=4B(2), pad_enable, pad_interval=6 (128 dwords),
            // pad_amount=7 (8 dwords)  -> LDS row pitch = 136 floats
            g1[0] = (2 << 16) | (1 << 20) | (6 << 22) | (7 << 25);
            g1[1] = (HW & 0xffff) << 16;                  // tensor_dim0 = 256
            g1[2] = (EMB_DIM & 0xffff) << 16;             // tensor_dim1 = 64
            g1[3] = (TILE_W & 0xffff) << 16;              // tile_dim0   = 128
            g1[4] = EMB_DIM;                              // tile_dim1=64, tile_dim2=0
            g1[5] = HW;                                   // tensor_dim0_stride = 256
            g1[6] = 0;
            g1[7] = 0;

            v4i g2 = {0, 0, 0, 0};
            v4i g3 = {0, 0, 0, 0};
#if __clang_major__ >= 23
            v8i g4 = {0, 0, 0, 0, 0, 0, 0, 0};
            __builtin_amdgcn_tensor_load_to_lds(g0, g1, g2, g3, g4, 0);
#else
            __builtin_amdgcn_tensor_load_to_lds(g0, g1, g2, g3, 0);
#endif
            __builtin_amdgcn_s_wait_tensorcnt(0);
        }
        __syncthreads();   // LDS tile visible to all waves

        // ---- Load A tile = -2*z from LDS, ISA 16x4-f32 operand layout ----
        // instr i, vgpr j, lane l: A[m=l%16, k=4i + 2*(l/16) + j]
        const int hwl = wave * 16 + l16;   // phase-local pixel column
        v2f a2[16];
#pragma unroll
        for (int i = 0; i < 16; ++i) {
            const int k0 = 4 * i + 2 * lh;
            float x0 = z_lds[k0 * ROW_PITCH + hwl];
            float x1 = z_lds[(k0 + 1) * ROW_PITCH + hwl];
            a2[i][0] = -2.0f * x0;
            a2[i][1] = -2.0f * x1;
        }

        float bestV[8];
        int   bestI[8];
#pragma unroll
        for (int j = 0; j < 8; ++j) { bestV[j] = 3.4e38f; bestI[j] = 0x7fffffff; }

        // ---- Sweep the 32 code tiles (16 codes each) ----
        for (int t = 0; t < 32; ++t) {
            const int n = t * 16 + l16;            // this lane's code column
            const float* erow = e + (size_t)n * EMB_DIM + 2 * lh;
            const float  cini = e2[n];
            v8f c = {cini, cini, cini, cini, cini, cini, cini, cini};
#pragma unroll
            for (int i = 0; i < 16; ++i) {
                // B operand: lane l, vgpr j: B[k=4i+2*(l/16)+j, n=l%16]
                v2f bb = *(const v2f*)(erow + 4 * i);
                c = __builtin_amdgcn_wmma_f32_16x16x4_f32(
                        false, a2[i], false, bb, (short)0, c, false, false);
            }
            // C layout: vgpr j, lane l -> (M = j + 8*(l/16), N = l%16)
#pragma unroll
            for (int j = 0; j < 8; ++j) {
                float v = c[j];
                if (v < bestV[j]) { bestV[j] = v; bestI[j] = n; }
            }
        }

        // ---- Reduce argmin across the 16 columns of each half-wave ----
        // (tie -> smaller code index, matching jnp.argmin first-occurrence)
#pragma unroll
        for (int j = 0; j < 8; ++j) {
            float v  = bestV[j];
            int   ii = bestI[j];
#pragma unroll
            for (int m = 1; m <= 8; m <<= 1) {
                float ov = __shfl_xor(v, m, 32);
                int   oi = __shfl_xor(ii, m, 32);
                if (ov < v || (ov == v && oi < ii)) { v = ov; ii = oi; }
            }
            bestI[j] = ii;
        }

        if (l16 == 0) {
            // lane 0 holds rows 0..7 of the tile, lane 16 holds rows 8..15
            const int hw0 = p * TILE_W + wave * 16;
#pragma unroll
            for (int j = 0; j < 8; ++j) {
                const int r = j + 8 * lh;
                idx_s[hw0 + r] = (unsigned short)bestI[j];
                atomicAdd(&hist[bestI[j]], 1u);          // ds_add_u32
            }
        }
        __syncthreads();   // done reading z_lds before next TDM overwrite
    }

    // ---- Gather: thread tid owns pixel hw=tid; coalesced 1KB store per d ----
    const int code = idx_s[tid];
    const float4* er = (const float4*)(e + (size_t)code * EMB_DIM);
    float* out = z_q + (size_t)b * IMG_ELEMS + tid;
#pragma unroll
    for (int d4 = 0; d4 < EMB_DIM / 4; ++d4) {
        float4 v = er[d4];                                // global_load_b128
        out[(4 * d4 + 0) * HW] = v.x;
        out[(4 * d4 + 1) * HW] = v.y;
        out[(4 * d4 + 2) * HW] = v.z;
        out[(4 * d4 + 3) * HW] = v.w;
    }

    // ---- Flush block histogram to global (integer adds: deterministic) ----
    unsigned int h0 = hist[tid];
    unsigned int h1 = hist[tid + 256];
    if (h0) atomicAdd(&n_i[tid], h0);
    if (h1) atomicAdd(&n_i[tid + 256], h1);
}

// ---------------------------------------------------------------------------
// Kernel 2: N_new = decay*N + (1-decay)*n_i
// ---------------------------------------------------------------------------
__global__ void vq_ema_kernel(const float* __restrict__ N,
                              const unsigned int* __restrict__ n_i,
                              float* __restrict__ N_out) {
    int k = blockIdx.x * blockDim.x + threadIdx.x;
    if (k < NUM_EMB) {
        N_out[k] = 0.995f * N[k] + 0.005f * (float)n_i[k];
    }
}

extern "C" void kernel_launch(void* const* d_in, const int* in_sizes, int n_in,
                              void* d_out, int out_size, void* d_ws, size_t ws_size,
                              hipStream_t stream) {
    const float* z_e = (const float*)d_in[0];   // (1024, 64, 16, 16) f32
    const float* e   = (const float*)d_in[1];   // (512, 64) f32
    const float* N   = (const float*)d_in[2];   // (512,) f32

    float* z_q   = (float*)d_out;               // first 16,777,216 floats
    float* N_out = (float*)d_out + ZQ_ELEMS;    // next 512 floats

    float*        e2  = (float*)d_ws;                          // 512 f32
    unsigned int* n_i = (unsigned int*)((char*)d_ws + 2048);   // 512 u32

    vq_prep_kernel<<<2, 256, 0, stream>>>(e, e2, n_i);
    vq_main_kernel<<<1024, 256, 0, stream>>>(z_e, e, e2, z_q, n_i);
    vq_ema_kernel<<<2, 256, 0, stream>>>(N, n_i, N_out);
}